// PCALayer_13950053777850
// MI455X (gfx1250) — compile-verified
//
#include <hip/hip_runtime.h>

typedef __attribute__((ext_vector_type(2))) float v2f;
typedef __attribute__((ext_vector_type(8))) float v8f;

#define B_  16
#define N_  65536
#define C_  64
#define K_  32

// ---- workspace layout (floats) ----
// [0,1024)        : per-batch channel sums   (B*64)
// [1024,66560)    : per-batch gram matrices  (B*64*64)
// [66560,99328)   : per-batch top eigvecs    (B*64*32), row-major [c][k]
// [99328,99840)   : per-batch projection offsets mu^T*W (B*32)
#define OFF_SUMS 0
#define OFF_GRAM 1024
#define OFF_TOP  66560
#define OFF_OFFS 99328
#define ZERO_N   OFF_TOP

// ---------------- stage 0: zero the accumulators ----------------
__global__ void pca_zero(float* __restrict__ ws) {
    int i = blockIdx.x * 256 + threadIdx.x;
    if (i < ZERO_N) ws[i] = 0.0f;
}

// ---------------- stage 1: sums + X^T X gram via f32 WMMA ----------------
// grid: B_ * SCHUNKS blocks of 256 threads (8 waves). Waves start 128 apart; each reads 512
// samples. Gram is symmetric: only the 10 upper-triangular 16x16 tiles are
// accumulated in registers (80 VGPRs), mirrored during the LDS reduction.
#define SCHUNKS 64
#define CHUNK   (N_ / SCHUNKS)      // 1024 samples per workgroup

__global__ __launch_bounds__(256) void pca_gram(const float* __restrict__ x,
                                                float* __restrict__ ws) {
    const int b     = blockIdx.x / SCHUNKS;
    const int chunk = blockIdx.x % SCHUNKS;
    const int wave  = threadIdx.x >> 5;
    const int lane  = threadIdx.x & 31;
    const int lm    = lane & 15;     // lane % 16
    const int lh    = lane >> 4;     // 0 or 1

    const float* xb = x + (size_t)b * N_ * C_;
    const int s0 = chunk * CHUNK + wave * (CHUNK / 8);   // 128 samples/wave

    v8f acc[10] = {};
    float sumacc[4] = {0.f, 0.f, 0.f, 0.f};

    for (int it = 0; it < CHUNK / 8; ++it) {             // 4 samples per step
        const int s = s0 + 4 * it;
        const float* p0 = xb + (size_t)(s + 2 * lh) * C_;
        __builtin_prefetch(p0 + 16 * C_, 0, 1);          // ~4KB ahead
        v2f f[4];
#pragma unroll
        for (int c = 0; c < 4; ++c) {
            v2f t;
            t.x = p0[c * 16 + lm];            // sample s+2lh,   K = v + 2*laneHi
            t.y = p0[C_ + c * 16 + lm];       // sample s+2lh+1
            f[c] = t;
            sumacc[c] += t.x + t.y;
        }
        // upper-triangular tile grid: (i,j), i<=j
#pragma unroll
        for (int t = 0; t < 10; ++t) {
            constexpr int TI[10] = {0, 0, 0, 0, 1, 1, 1, 2, 2, 3};
            constexpr int TJ[10] = {0, 1, 2, 3, 1, 2, 3, 2, 3, 3};
            acc[t] = __builtin_amdgcn_wmma_f32_16x16x4_f32(
                false, f[TI[t]], false, f[TJ[t]], (short)0, acc[t], false, false);
        }
    }

    // reduce 8 waves -> LDS (ds_add_f32), mirroring i<j tiles, then one
    // atomic pass to global gram.
    __shared__ float lgram[64 * 64];
    for (int i = threadIdx.x; i < 4096; i += 256) lgram[i] = 0.0f;
    __syncthreads();
#pragma unroll
    for (int t = 0; t < 10; ++t) {
        constexpr int TI[10] = {0, 0, 0, 0, 1, 1, 1, 2, 2, 3};
        constexpr int TJ[10] = {0, 1, 2, 3, 1, 2, 3, 2, 3, 3};
        const int ti = TI[t], tj = TJ[t];
#pragma unroll
        for (int r = 0; r < 8; ++r) {
            const int row = ti * 16 + r + 8 * lh;   // C/D layout: M = r + 8*laneHi
            const int col = tj * 16 + lm;           // N = lane % 16
            atomicAdd(&lgram[row * 64 + col], acc[t][r]);
            if (ti != tj) atomicAdd(&lgram[col * 64 + row], acc[t][r]);
        }
    }
    __syncthreads();

    float* gram = ws + OFF_GRAM + b * 4096;
    for (int i = threadIdx.x; i < 4096; i += 256) atomicAdd(&gram[i], lgram[i]);

    float* sums = ws + OFF_SUMS + b * 64;
#pragma unroll
    for (int c = 0; c < 4; ++c) atomicAdd(&sums[c * 16 + lm], sumacc[c]);
}

// ---------------- stage 2: covariance -> Jacobi eigh -> top-32 ----------------
// one 256-thread workgroup per batch; A and V live in LDS.
__global__ void pca_eigh(float* __restrict__ ws) {
    __shared__ float A[64 * 64];
    __shared__ float V[64 * 64];
    __shared__ float smean[64];
    __shared__ int   pp[32], pq[32];
    __shared__ float pc[32], ps[32];
    __shared__ float eval[64];
    __shared__ int   idx[64];

    const int b = blockIdx.x;
    const int t = threadIdx.x;
    const float* sums = ws + OFF_SUMS + b * 64;
    const float* gram = ws + OFF_GRAM + b * 4096;
    float* topw  = ws + OFF_TOP  + b * 2048;
    float* offsw = ws + OFF_OFFS + b * 32;

    if (t < 64) smean[t] = sums[t] * (1.0f / (float)N_);
    __syncthreads();

    const float invNm1 = 1.0f / (float)(N_ - 1);
    for (int i = t; i < 4096; i += 256) {
        const int r = i >> 6, c = i & 63;
        A[i] = (gram[i] - (float)N_ * smean[r] * smean[c]) * invNm1;
        V[i] = (r == c) ? 1.0f : 0.0f;
    }
    __syncthreads();

    // cyclic Jacobi, round-robin pairing: 63 rounds x 32 disjoint pairs
    for (int sweep = 0; sweep < 12; ++sweep) {
        for (int rr = 0; rr < 63; ++rr) {
            if (t < 32) {
                int p = (t == 0) ? 0 : 1 + ((t - 1 + rr) % 63);
                int q = 1 + ((62 - t + rr) % 63);
                if (p > q) { int tmp = p; p = q; q = tmp; }
                const float apq = A[p * 64 + q];
                float c, s;
                if (fabsf(apq) < 1e-20f) { c = 1.0f; s = 0.0f; }
                else {
                    const float tau = (A[q * 64 + q] - A[p * 64 + p]) / (2.0f * apq);
                    const float tt  = ((tau >= 0.0f) ? 1.0f : -1.0f) /
                                      (fabsf(tau) + sqrtf(1.0f + tau * tau));
                    c = 1.0f / sqrtf(1.0f + tt * tt);
                    s = tt * c;
                }
                pp[t] = p; pq[t] = q; pc[t] = c; ps[t] = s;
            }
            __syncthreads();
            // A <- A*J  and  V <- V*J   (column rotations)
            for (int u = t; u < 2048; u += 256) {
                const int pr = u >> 6, i = u & 63;
                const int p = pp[pr], q = pq[pr];
                const float c = pc[pr], s = ps[pr];
                const float aip = A[i * 64 + p], aiq = A[i * 64 + q];
                A[i * 64 + p] = c * aip - s * aiq;
                A[i * 64 + q] = s * aip + c * aiq;
                const float vip = V[i * 64 + p], viq = V[i * 64 + q];
                V[i * 64 + p] = c * vip - s * viq;
                V[i * 64 + q] = s * vip + c * viq;
            }
            __syncthreads();
            // A <- J^T * A   (row rotations)
            for (int u = t; u < 2048; u += 256) {
                const int pr = u >> 6, j = u & 63;
                const int p = pp[pr], q = pq[pr];
                const float c = pc[pr], s = ps[pr];
                const float apj = A[p * 64 + j], aqj = A[q * 64 + j];
                A[p * 64 + j] = c * apj - s * aqj;
                A[q * 64 + j] = s * apj + c * aqj;
            }
            __syncthreads();
        }
    }

    if (t == 0) {
        for (int i = 0; i < 64; ++i) { eval[i] = A[i * 64 + i]; idx[i] = i; }
        for (int i = 0; i < 32; ++i) {          // partial selection sort, top-32
            int best = i;
            for (int j = i + 1; j < 64; ++j)
                if (eval[idx[j]] > eval[idx[best]]) best = j;
            const int tmp = idx[i]; idx[i] = idx[best]; idx[best] = tmp;
        }
    }
    __syncthreads();
    for (int u = t; u < 2048; u += 256) {
        const int c = u >> 5, k = u & 31;
        topw[u] = V[c * 64 + idx[k]];          // top[c][k], row-major 64x32
    }
    // projection offsets: offs[k] = mu^T * W[:,k]  (reads only smean/V/idx)
    if (t < 32) {
        const int col = idx[t];
        float a = 0.0f;
        for (int c = 0; c < 64; ++c) a += smean[c] * V[c * 64 + col];
        offsw[t] = a;
    }
}

// ---------------- stage 3: projection via f32 WMMA ----------------
// out = x*W - (mu^T W); grid: B_ * PT blocks of 256 threads.
#define PT 128
#define PCHUNK (N_ / PT)                 // 512 samples per workgroup

__global__ __launch_bounds__(256) void pca_proj(const float* __restrict__ x,
                                                const float* __restrict__ ws,
                                                float* __restrict__ out) {
    const int b      = blockIdx.x / PT;
    const int cchunk = blockIdx.x % PT;
    const int wave   = threadIdx.x >> 5;
    const int lane   = threadIdx.x & 31;
    const int lm     = lane & 15;
    const int lh     = lane >> 4;

    const float* xb    = x  + (size_t)b * N_ * C_;
    const float* topw  = ws + OFF_TOP  + b * 2048;
    const float* offsw = ws + OFF_OFFS + b * 32;

    // preload all B fragments (64x32 eigvec block = 64 VGPRs)
    v2f bf[32];
#pragma unroll
    for (int kk = 0; kk < 16; ++kk) {
        const int ch = 4 * kk + 2 * lh;          // B layout: K = v + 2*laneHi
#pragma unroll
        for (int j = 0; j < 2; ++j) {
            v2f bb;
            bb.x = topw[(ch)     * 32 + 16 * j + lm];
            bb.y = topw[(ch + 1) * 32 + 16 * j + lm];
            bf[kk * 2 + j] = bb;
        }
    }
    const float o0 = offsw[lm];          // comp lm
    const float o1 = offsw[16 + lm];     // comp 16+lm

    const int base = cchunk * PCHUNK + wave * (PCHUNK / 8);   // 64 samples/wave
#pragma unroll
    for (int tile = 0; tile < PCHUNK / 8 / 16; ++tile) {      // 4 tiles of 16
        const int s0 = base + tile * 16;
        const float* xs = xb + (size_t)s0 * C_;
        __builtin_prefetch(xs + 16 * C_ + lane * 32, 0, 1);   // next tile
        v8f acc0 = {}, acc1 = {};
#pragma unroll
        for (int kk = 0; kk < 16; ++kk) {
            // A layout: M = lane%16 (sample), K = v + 2*laneHi; 8B aligned
            const v2f a = *(const v2f*)(xs + (size_t)lm * C_ + 4 * kk + 2 * lh);
            acc0 = __builtin_amdgcn_wmma_f32_16x16x4_f32(
                false, a, false, bf[kk * 2 + 0], (short)0, acc0, false, false);
            acc1 = __builtin_amdgcn_wmma_f32_16x16x4_f32(
                false, a, false, bf[kk * 2 + 1], (short)0, acc1, false, false);
        }
        float* o = out + ((size_t)b * N_ + s0) * K_;
#pragma unroll
        for (int r = 0; r < 8; ++r) {
            const int row = r + 8 * lh;
            o[row * K_ + lm]      = acc0[r] - o0;
            o[row * K_ + 16 + lm] = acc1[r] - o1;
        }
    }
}

extern "C" void kernel_launch(void* const* d_in, const int* in_sizes, int n_in,
                              void* d_out, int out_size, void* d_ws, size_t ws_size,
                              hipStream_t stream) {
    const float* x = (const float*)d_in[0];
    float* out = (float*)d_out;
    float* ws  = (float*)d_ws;

    pca_zero<<<(ZERO_N + 255) / 256, 256, 0, stream>>>(ws);
    pca_gram<<<B_ * SCHUNKS, 256, 0, stream>>>(x, ws);
    pca_eigh<<<B_, 256, 0, stream>>>(ws);
    pca_proj<<<B_ * PT, 256, 0, stream>>>(x, ws, out);
}